// PSAMask_82343112999328
// MI455X (gfx1250) — compile-verified
//
#include <hip/hip_runtime.h>
#include <hip/hip_bf16.h>

// PSAMask "collect" specialized for N=2, H=W=59, mask_H=mask_W=117.
//   out[n, oh*W+ow, h, w] = x[n, (oh-h+half)*MW + (ow-w+half), h, w]
// (validity mask is always true for mask = 2H-1).
//
// One block per (n, oh, h); dh = oh-h+half fixed per block.
// Phase 1: async-DMA (ASYNCcnt) the trimmed diagonal band
//          x[n, dh*MW+dw, h, valid-w-range] into LDS (pitch 60 -> LDS lane
//          stride -59 ≡ 5 mod 64, conflict-free).
// Phase 2: async-DMA contiguous 59-float output rows straight from LDS to
//          global (no VGPR round-trip).
// Memory-bound: ~194 MB total -> ~8.3 us at 23.3 TB/s.

#define MH   117
#define MW   117
#define H_   59
#define W_   59
#define NB   2
#define HALF ((MW - 1) / 2)      // 58
#define CIN  (MH * MW)           // 13689
#define HW   (H_ * W_)           // 3481
#define PITCH 60

#if defined(__has_builtin)
#  if __has_builtin(__builtin_amdgcn_global_load_async_to_lds_b32)
#    define HAVE_ASYNC_LD 1
#  endif
#  if __has_builtin(__builtin_amdgcn_global_store_async_from_lds_b32)
#    define HAVE_ASYNC_ST 1
#  endif
#  if __has_builtin(__builtin_amdgcn_s_wait_asynccnt)
#    define HAVE_WAIT_ASYNC 1
#  endif
#endif

#if defined(HAVE_WAIT_ASYNC)
#  define WAIT_ASYNC0() __builtin_amdgcn_s_wait_asynccnt(0)
#else
#  define WAIT_ASYNC0() asm volatile("s_wait_asynccnt 0" ::: "memory")
#endif

typedef __attribute__((address_space(1))) int* gip;   // global-AS int* (builtin param type)
typedef __attribute__((address_space(3))) int* lip;   // LDS-AS int*

__global__ __launch_bounds__(256) void psamask_collect_kernel(
    const float* __restrict__ x, float* __restrict__ out) {
  __shared__ float tile[MW * PITCH];   // 117*60*4 = 28,080 B

  const int h   = blockIdx.x;   // sp_h
  const int oh  = blockIdx.y;
  const int n   = blockIdx.z;
  const int tid = threadIdx.x;

  const int dh = oh - h + HALF;        // always in [0, MH) for this shape

  // x[n, dh*MW + dw, h, w] = src[dw*HW + w]
  const float* src = x + (((size_t)n * CIN + (size_t)dh * MW) * H_ + h) * W_;

  // ---- Phase 1: trimmed diagonal band -> LDS via async global->LDS ----
  // Entry (dw, w) is needed iff ow = dw + w - HALF is in [0, W_).
  for (int idx = tid; idx < MW * W_; idx += 256) {
    const int dw = idx / W_;
    const int w  = idx - dw * W_;
    const int ow = dw + w - HALF;
    if (ow >= 0 && ow < W_) {
#if defined(HAVE_ASYNC_LD)
      __builtin_amdgcn_global_load_async_to_lds_b32(
          (gip)(void*)(src + (size_t)dw * HW + w),
          (lip)(void*)&tile[dw * PITCH + w],
          /*offset=*/0, /*cpol=*/0);
#else
      tile[dw * PITCH + w] = src[(size_t)dw * HW + w];
#endif
    }
  }

#if defined(HAVE_ASYNC_LD)
  WAIT_ASYNC0();
#endif
  __syncthreads();

  // ---- Phase 2: coalesced output rows straight from LDS via async store ----
  // out[((n*HW + oh*W_ + ow)*H_ + h)*W_ + w] = tile[(ow - w + HALF)*PITCH + w]
  float* dst = out + (((size_t)n * HW + (size_t)oh * W_) * H_ + h) * W_;
  for (int idx = tid; idx < HW; idx += 256) {
    const int ow = idx / W_;
    const int w  = idx - ow * W_;
    const int dw = ow - w + HALF;      // in [0, MW) by construction
#if defined(HAVE_ASYNC_ST)
    __builtin_amdgcn_global_store_async_from_lds_b32(
        (gip)(void*)(dst + (size_t)ow * HW + w),
        (lip)(void*)&tile[dw * PITCH + w],
        /*offset=*/0, /*cpol=*/0);
#else
    dst[(size_t)ow * HW + w] = tile[dw * PITCH + w];
#endif
  }

#if defined(HAVE_ASYNC_ST)
  WAIT_ASYNC0();   // drain async stores before wave termination
#endif
}

extern "C" void kernel_launch(void* const* d_in, const int* in_sizes, int n_in,
                              void* d_out, int out_size, void* d_ws, size_t ws_size,
                              hipStream_t stream) {
  (void)in_sizes; (void)n_in; (void)out_size; (void)d_ws; (void)ws_size;
  const float* x = (const float*)d_in[0];
  float* out = (float*)d_out;
  dim3 grid(H_, H_, NB);   // (h, oh, n)
  psamask_collect_kernel<<<grid, 256, 0, stream>>>(x, out);
}